// EGAT_2L_22539988370185
// MI455X (gfx1250) — compile-verified
//
#include <hip/hip_runtime.h>
#include <cstddef>

typedef float v2f __attribute__((ext_vector_type(2)));
typedef float v8f __attribute__((ext_vector_type(8)));

#define NEG_SLOPE 0.2f

// ---------- helpers: ordered-uint encoding for float atomic max ----------
__device__ __forceinline__ unsigned fenc(float f) {
    unsigned u = __float_as_uint(f);
    return (u >> 31) ? ~u : (u | 0x80000000u);
}
__device__ __forceinline__ float fdec(unsigned e) {
    unsigned u = (e >> 31) ? (e & 0x7fffffffu) : ~e;
    return __uint_as_float(u);
}
// fenc(-inf) == 0x007FFFFF
#define ENC_NEG_INF 0x007FFFFFu

// ---------- small init / reduction kernels ----------
__global__ void k_init_misc(float* misc) {
    if (threadIdx.x < 128) misc[threadIdx.x] = 0.f;
}

// column-sum of edge_attr [E,16] into misc[0..15]
__global__ void k_mean_reduce(const float* __restrict__ eattr, float* __restrict__ acc, int E) {
    __shared__ float sm[16];
    if (threadIdx.x < 16) sm[threadIdx.x] = 0.f;
    __syncthreads();
    float loc[16];
#pragma unroll
    for (int j = 0; j < 16; ++j) loc[j] = 0.f;
    for (int r = blockIdx.x * blockDim.x + threadIdx.x; r < E; r += gridDim.x * blockDim.x) {
        const float* row = eattr + (size_t)r * 16;
#pragma unroll
        for (int j = 0; j < 16; ++j) loc[j] += row[j];
    }
#pragma unroll
    for (int j = 0; j < 16; ++j) atomicAdd(&sm[j], loc[j]);
    __syncthreads();
    if (threadIdx.x < 16) atomicAdd(&acc[threadIdx.x], sm[threadIdx.x]);
}

// misc[0..15]  : mean_e   (in: column sums)
// misc[16..31] : we_a1 = We1 @ a_edge1
// misc[32..47] : we_a2 = We2 @ a_edge2
// misc[48]     : selfc1 = mean_e . we_a1   (self-loop edge logit, layer 1)
// misc[49]     : selfc2
__global__ void k_prep(float* misc, const float* We1, const float* ae1,
                       const float* We2, const float* ae2, int E) {
    int t = threadIdx.x;
    if (t < 16) {
        misc[t] = misc[t] / (float)E;
        float w1 = 0.f, w2 = 0.f;
        for (int j = 0; j < 64; ++j) {
            w1 += We1[t * 64 + j] * ae1[j];
            w2 += We2[t * 64 + j] * ae2[j];
        }
        misc[16 + t] = w1;
        misc[32 + t] = w2;
    }
    __syncthreads();
    if (t == 0) {
        float c1 = 0.f, c2 = 0.f;
        for (int d = 0; d < 16; ++d) {
            c1 += misc[d] * misc[16 + d];
            c2 += misc[d] * misc[32 + d];
        }
        misc[48] = c1;
        misc[49] = c2;
    }
}

__global__ void k_init_accum(float* __restrict__ out_un, float* __restrict__ denom,
                             unsigned* __restrict__ maxbuf, int n) {
    int i = blockIdx.x * blockDim.x + threadIdx.x;
    if (i < n * 64) out_un[i] = 0.f;
    if (i < n) { denom[i] = 0.f; maxbuf[i] = ENC_NEG_INF; }
}

// ---------- WMMA GEMM:  Y[nrows,64] = X[nrows,64] @ W[64,64]  (fp32) ----------
// 256 threads = 8 waves; wave w handles 16 rows via V_WMMA_F32_16X16X4_F32.
// Out-of-range rows are CLAMPED on the load side (values harmless, rows never
// stored) so the unrolled K-loop carries no per-lane EXEC predication.
__global__ void k_gemm64_wmma(const float* __restrict__ X, const float* __restrict__ W,
                              float* __restrict__ Y, int nrows) {
    int wave = threadIdx.x >> 5;
    int lane = threadIdx.x & 31;
    int row0 = (blockIdx.x * 8 + wave) * 16;
    if (row0 >= nrows) return;                 // wave-uniform exit, EXEC stays all-1s

    int r    = lane & 15;                      // A: row-in-tile;  B: col-in-tile
    int koff = (lane >> 4) * 2;                // K sub-offset per half-wave
    int arow = row0 + r;
    if (arow >= nrows) arow = nrows - 1;       // clamp instead of predicate
    const float* xr = X + (size_t)arow * 64 + koff;   // 8B-aligned (koff even)

    v8f c0 = {}, c1 = {}, c2 = {}, c3 = {};
#pragma unroll
    for (int k = 0; k < 64; k += 4) {
        v2f a = *(const v2f*)(xr + k);                 // global_load_b64
        const float* Wr0 = W + (size_t)(k + koff) * 64;
        const float* Wr1 = Wr0 + 64;
        v2f b0, b1, b2, b3;
        b0.x = Wr0[r];      b0.y = Wr1[r];
        b1.x = Wr0[r + 16]; b1.y = Wr1[r + 16];
        b2.x = Wr0[r + 32]; b2.y = Wr1[r + 32];
        b3.x = Wr0[r + 48]; b3.y = Wr1[r + 48];
        c0 = __builtin_amdgcn_wmma_f32_16x16x4_f32(false, a, false, b0, (short)0, c0, false, false);
        c1 = __builtin_amdgcn_wmma_f32_16x16x4_f32(false, a, false, b1, (short)0, c1, false, false);
        c2 = __builtin_amdgcn_wmma_f32_16x16x4_f32(false, a, false, b2, (short)0, c2, false, false);
        c3 = __builtin_amdgcn_wmma_f32_16x16x4_f32(false, a, false, b3, (short)0, c3, false, false);
    }
    // C layout: vgpr v -> row (v + 8*(lane/16)), col lane%16 (+16 per tile)
    int crow = row0 + (lane >> 4) * 8;
    int ccol = lane & 15;
#pragma unroll
    for (int v = 0; v < 8; ++v) {
        int rr = crow + v;
        if (rr < nrows) {
            float* yr = Y + (size_t)rr * 64 + ccol;
            yr[0]  = c0[v];
            yr[16] = c1[v];
            yr[32] = c2[v];
            yr[48] = c3[v];
        }
    }
}

// ---------- per-node attention dots: s_src = xp.a_src, s_dst = xp.a_dst ----------
__global__ void k_node_dots(const float* __restrict__ xp, const float* __restrict__ a_src,
                            const float* __restrict__ a_dst, float* __restrict__ s_src,
                            float* __restrict__ s_dst, int n) {
    int gw   = (blockIdx.x * blockDim.x + threadIdx.x) >> 5;   // one wave per node
    int lane = threadIdx.x & 31;
    if (gw >= n) return;
    float2 v  = ((const float2*)(xp + (size_t)gw * 64))[lane];
    float2 as = ((const float2*)a_src)[lane];
    float2 ad = ((const float2*)a_dst)[lane];
    float ss = v.x * as.x + v.y * as.y;
    float sd = v.x * ad.x + v.y * ad.y;
#pragma unroll
    for (int off = 16; off > 0; off >>= 1) {
        ss += __shfl_xor(ss, off, 32);
        sd += __shfl_xor(sd, off, 32);
    }
    if (lane == 0) { s_src[gw] = ss; s_dst[gw] = sd; }
}

// ---------- edge logits (leaky-relu) + segment max ----------
__global__ void k_edge_alpha_max(const long long* __restrict__ src, const long long* __restrict__ dst,
                                 const float* __restrict__ eattr, const float* __restrict__ we_a,
                                 const float* __restrict__ selfc_p,
                                 const float* __restrict__ s_src, const float* __restrict__ s_dst,
                                 float* __restrict__ alpha, unsigned* __restrict__ maxbuf,
                                 int E, int n) {
    int e = blockIdx.x * blockDim.x + threadIdx.x;
    int Etot = E + n;
    if (e >= Etot) return;
    int s, d; float ea;
    if (e < E) {
        s = (int)src[e]; d = (int)dst[e];
        const float* row = eattr + (size_t)e * 16;
        ea = 0.f;
#pragma unroll
        for (int j = 0; j < 16; ++j) ea += row[j] * we_a[j];
    } else {                                   // self-loop, edge_attr = mean
        s = d = e - E;
        ea = selfc_p[0];
    }
    float a = s_src[s] + s_dst[d] + ea;
    a = (a > 0.f) ? a : a * NEG_SLOPE;
    alpha[e] = a;
    atomicMax(maxbuf + d, fenc(a));
}

// ---------- exp + unnormalized scatter-aggregate (fused softmax-den + agg) ----------
__global__ void k_edge_aggregate(const long long* __restrict__ src, const long long* __restrict__ dst,
                                 const float* __restrict__ alpha, const unsigned* __restrict__ maxbuf,
                                 const float* __restrict__ xp, float* __restrict__ denom,
                                 float* __restrict__ out_un, int E, int n) {
    int gw   = (blockIdx.x * blockDim.x + threadIdx.x) >> 5;   // one wave per edge
    int lane = threadIdx.x & 31;
    int Etot = E + n;
    if (gw >= Etot) return;
    int s, d;
    if (gw < E) { s = (int)src[gw]; d = (int)dst[gw]; } else { s = d = gw - E; }
    float m  = fdec(maxbuf[d]);
    float ex = __expf(alpha[gw] - m);
    if (lane == 0) atomicAdd(denom + d, ex);
    float2 v = ((const float2*)(xp + (size_t)s * 64))[lane];
    float* o = out_un + (size_t)d * 64 + lane * 2;
    atomicAdd(o,     ex * v.x);
    atomicAdd(o + 1, ex * v.y);
}

// ---------- normalize + bias + relu ----------
__global__ void k_finalize(const float* __restrict__ out_un, const float* __restrict__ denom,
                           const float* __restrict__ b, float* __restrict__ out, int n) {
    int i = blockIdx.x * blockDim.x + threadIdx.x;
    if (i >= n * 64) return;
    float v = out_un[i] / (denom[i >> 6] + 1e-16f) + b[i & 63];
    out[i] = v > 0.f ? v : 0.f;
}

extern "C" void kernel_launch(void* const* d_in, const int* in_sizes, int n_in,
                              void* d_out, int out_size, void* d_ws, size_t ws_size,
                              hipStream_t stream) {
    (void)n_in; (void)out_size; (void)ws_size;
    const float*     x     = (const float*)d_in[0];
    const long long* ei    = (const long long*)d_in[1];   // int64 [2,E]
    const float*     eattr = (const float*)d_in[2];
    const float*     W1    = (const float*)d_in[3];
    const float*     as1   = (const float*)d_in[4];
    const float*     ad1   = (const float*)d_in[5];
    const float*     We1   = (const float*)d_in[6];
    const float*     ae1   = (const float*)d_in[7];
    const float*     b1    = (const float*)d_in[8];
    const float*     W2    = (const float*)d_in[9];
    const float*     as2   = (const float*)d_in[10];
    const float*     ad2   = (const float*)d_in[11];
    const float*     We2   = (const float*)d_in[12];
    const float*     ae2   = (const float*)d_in[13];
    const float*     b2    = (const float*)d_in[14];

    const int n    = in_sizes[0] / 64;    // nodes
    const int E    = in_sizes[2] / 16;    // edges
    const int Etot = E + n;               // edges incl. self loops
    const long long* src = ei;
    const long long* dst = ei + E;

    // workspace layout
    float* p    = (float*)d_ws;
    float* xp   = p; p += (size_t)n * 64;   // projected features (reused per layer)
    float* h    = p; p += (size_t)n * 64;   // layer-1 output
    float* oun  = p; p += (size_t)n * 64;   // unnormalized aggregate
    float* ssrc = p; p += n;
    float* sdst = p; p += n;
    float* den  = p; p += n;
    unsigned* mx = (unsigned*)p; p += n;
    float* alpha = p; p += Etot;
    float* misc  = p; p += 128;

    const int gemmBlocks = (n + 127) / 128;
    const int dotBlocks  = (int)(((size_t)n * 32 + 255) / 256);
    const int accBlocks  = (int)(((size_t)n * 64 + 255) / 256);
    const int eBlocks    = (Etot + 255) / 256;
    const int aggBlocks  = (int)(((long long)Etot * 32 + 255) / 256);

    k_init_misc<<<1, 128, 0, stream>>>(misc);
    k_mean_reduce<<<256, 256, 0, stream>>>(eattr, misc, E);
    k_prep<<<1, 64, 0, stream>>>(misc, We1, ae1, We2, ae2, E);

    // ---- layer 1 ----
    k_gemm64_wmma<<<gemmBlocks, 256, 0, stream>>>(x, W1, xp, n);
    k_node_dots<<<dotBlocks, 256, 0, stream>>>(xp, as1, ad1, ssrc, sdst, n);
    k_init_accum<<<accBlocks, 256, 0, stream>>>(oun, den, mx, n);
    k_edge_alpha_max<<<eBlocks, 256, 0, stream>>>(src, dst, eattr, misc + 16, misc + 48,
                                                  ssrc, sdst, alpha, mx, E, n);
    k_edge_aggregate<<<aggBlocks, 256, 0, stream>>>(src, dst, alpha, mx, xp, den, oun, E, n);
    k_finalize<<<accBlocks, 256, 0, stream>>>(oun, den, b1, h, n);

    // ---- layer 2 ----
    k_gemm64_wmma<<<gemmBlocks, 256, 0, stream>>>(h, W2, xp, n);
    k_node_dots<<<dotBlocks, 256, 0, stream>>>(xp, as2, ad2, ssrc, sdst, n);
    k_init_accum<<<accBlocks, 256, 0, stream>>>(oun, den, mx, n);
    k_edge_alpha_max<<<eBlocks, 256, 0, stream>>>(src, dst, eattr, misc + 32, misc + 49,
                                                  ssrc, sdst, alpha, mx, E, n);
    k_edge_aggregate<<<aggBlocks, 256, 0, stream>>>(src, dst, alpha, mx, xp, den, oun, E, n);
    k_finalize<<<accBlocks, 256, 0, stream>>>(oun, den, b2, (float*)d_out, n);
}